// RoiPoolingConv_15564961480841
// MI455X (gfx1250) — compile-verified
//
#include <hip/hip_runtime.h>
#include <cstdint>

#define ROW_POOL 7
#define COL_POOL 7
#define NUM_ROIS 300
#define IMG_H    96
#define IMG_W    160
#define IMG_C    1024

typedef float v4f __attribute__((ext_vector_type(4)));

// tf.image.resize-style half-pixel-center bilinear coords for one axis.
__device__ __forceinline__ void axis_coords(int start, int size, int i, int out_size,
                                            int* lo, int* hi, float* t) {
    const float sizef = (float)size;
    const float src   = ((float)i + 0.5f) * (sizef / (float)out_size) - 0.5f;
    const float f     = floorf(src);
    const float hic   = (float)(size - 1);
    const float flo   = fminf(fmaxf(f, 0.0f), hic);
    const float fhi   = fminf(fmaxf(f + 1.0f, 0.0f), hic);
    *lo = (int)flo + start;
    *hi = (int)fhi + start;
    *t  = src - f;
}

// One block per (roi, out_row). 256 threads; thread t owns channels [4t, 4t+4).
// Corners for output column c are staged into LDS with
// global_load_async_to_lds_b128 (ASYNCcnt), double-buffered so column c+1's
// gather overlaps column c's lerp + store. Each lane reads back only the 16
// bytes it loaded itself -> per-wave s_wait_asynccnt is the only sync needed.
__global__ __launch_bounds__(256) void roi_align_async_kernel(
    const float* __restrict__ img,   // (H, W, C) fp32
    const int*   __restrict__ rois,  // (NUM_ROIS, 4) int32 as (x, y, w, h)
    float*       __restrict__ out)   // (NUM_ROIS, 7, 7, C) fp32
{
    __shared__ __align__(16) float lds[2][4][IMG_C];   // 32 KB: 2 bufs x 4 corners

    const int blk = blockIdx.x;
    const int roi = blk / ROW_POOL;
    const int r   = blk % ROW_POOL;
    const int tid = threadIdx.x;

    const int bx = rois[roi * 4 + 0];
    const int by = rois[roi * 4 + 1];
    const int bw = rois[roi * 4 + 2];
    const int bh = rois[roi * 4 + 3];

    int lo_r, hi_r; float tr;
    axis_coords(by, bh, r, ROW_POOL, &lo_r, &hi_r, &tr);

    int lo_c[COL_POOL], hi_c[COL_POOL]; float tc[COL_POOL];
#pragma unroll
    for (int c = 0; c < COL_POOL; ++c)
        axis_coords(bx, bw, c, COL_POOL, &lo_c[c], &hi_c[c], &tc[c]);

    const size_t   row_lo = (size_t)lo_r * IMG_W;
    const size_t   row_hi = (size_t)hi_r * IMG_W;
    const unsigned ch     = (unsigned)tid * 4u;          // first of 4 channels

    // Issue the 4 corner b128 async loads for output column c into buffer buf.
    auto issue = [&](int c, int buf) {
        const float* p00 = img + (row_lo + (size_t)lo_c[c]) * IMG_C + ch;
        const float* p01 = img + (row_lo + (size_t)hi_c[c]) * IMG_C + ch;
        const float* p10 = img + (row_hi + (size_t)lo_c[c]) * IMG_C + ch;
        const float* p11 = img + (row_hi + (size_t)hi_c[c]) * IMG_C + ch;
        // Low 32 bits of the flat address of a __shared__ object are the LDS
        // byte offset (shared aperture base has zero low bits).
        const unsigned l00 = (unsigned)(uintptr_t)&lds[buf][0][ch];
        const unsigned l01 = (unsigned)(uintptr_t)&lds[buf][1][ch];
        const unsigned l10 = (unsigned)(uintptr_t)&lds[buf][2][ch];
        const unsigned l11 = (unsigned)(uintptr_t)&lds[buf][3][ch];
        asm volatile("global_load_async_to_lds_b128 %0, %1, off"
                     :: "v"(l00), "v"(p00) : "memory");
        asm volatile("global_load_async_to_lds_b128 %0, %1, off"
                     :: "v"(l01), "v"(p01) : "memory");
        asm volatile("global_load_async_to_lds_b128 %0, %1, off"
                     :: "v"(l10), "v"(p10) : "memory");
        asm volatile("global_load_async_to_lds_b128 %0, %1, off"
                     :: "v"(l11), "v"(p11) : "memory");
    };

    issue(0, 0);   // prime the pipeline

    float* out_row = out + (((size_t)roi * ROW_POOL + r) * COL_POOL) * IMG_C + ch;

#pragma unroll
    for (int c = 0; c < COL_POOL; ++c) {
        const int buf = c & 1;
        if (c + 1 < COL_POOL) {
            issue(c + 1, (c + 1) & 1);
            // 8 async loads outstanding; loads complete in order, so <=4
            // guarantees the 4 loads feeding buffer `buf` have landed.
            asm volatile("s_wait_asynccnt 0x4" ::: "memory");
        } else {
            asm volatile("s_wait_asynccnt 0x0" ::: "memory");
        }

        const v4f v00 = *(const v4f*)&lds[buf][0][ch];
        const v4f v01 = *(const v4f*)&lds[buf][1][ch];
        const v4f v10 = *(const v4f*)&lds[buf][2][ch];
        const v4f v11 = *(const v4f*)&lds[buf][3][ch];

        const v4f top = v00 + (v01 - v00) * tc[c];
        const v4f bot = v10 + (v11 - v10) * tc[c];
        const v4f res = top + (bot - top) * tr;

        *(v4f*)(out_row + (size_t)c * IMG_C) = res;
    }
}

extern "C" void kernel_launch(void* const* d_in, const int* in_sizes, int n_in,
                              void* d_out, int out_size, void* d_ws, size_t ws_size,
                              hipStream_t stream) {
    (void)in_sizes; (void)n_in; (void)out_size; (void)d_ws; (void)ws_size;
    const float* img  = (const float*)d_in[0];   // (1, 96, 160, 1024) fp32
    const int*   rois = (const int*)d_in[1];     // (1, 300, 4) int32
    float*       out  = (float*)d_out;           // (1, 300, 7, 7, 1024) fp32

    dim3 grid(NUM_ROIS * ROW_POOL);              // 2100 blocks
    dim3 block(256);                             // 8 wave32 per block
    roi_align_async_kernel<<<grid, block, 0, stream>>>(img, rois, out);
}